// GenLoss_85091892068580
// MI455X (gfx1250) — compile-verified
//
#include <hip/hip_runtime.h>

// ---------------------------------------------------------------------------
// GenLoss (masked per-channel MAE + adversarial term) for MI455X / gfx1250.
//
// Roofline: ~100.7 MB mandatory reads, O(1) output -> memory bound,
// floor ~4.3us at 23.3 TB/s. Strategy: coalesced float4 streaming loads,
// f32 accumulation, deterministic two-kernel reduction (no float atomics).
// Wave-level sum uses V_WMMA_F32_16X16X4_F32 (ones-matrix trick). Channel
// validity is tracked as a float max-|t| accumulator (clean v_max_num_f32,
// no i1 bit-packing), with a single compare + ballot at loop exit.
// ---------------------------------------------------------------------------

typedef __attribute__((ext_vector_type(2))) float v2f;
typedef __attribute__((ext_vector_type(8))) float v8f;

#define N_CH      48        // 16 batches * 3 channels
#define SLICES    32        // blocks per channel
#define CH_ELEMS  262144    // 512*512
#define SLICE_ELEMS (CH_ELEMS / SLICES)   // 8192 floats per block
#define NTHREADS  256       // 8 wave32
#define VEC_ITERS (SLICE_ELEMS / (NTHREADS * 4))  // 8 float4 per thread

// Full 32-lane sum via one f32 WMMA:
// A (16x4 f32): lanes 0-15 hold K=0 in VGPR0, lanes 16-31 hold K=2 in VGPR0
// (VGPR1 / K=1,3 zero). B = all ones => D[m,n] = p[m] + p[m+16].
// Each lane sums its 8 D-VGPRs (8 row-sums), then one shfl_xor(16) yields the
// full 32-lane total in every lane. Exact f32 FMA adds (RNE).
__device__ __forceinline__ float wave_sum_wmma(float x) {
    v2f a; a[0] = x;    a[1] = 0.0f;
    v2f b; b[0] = 1.0f; b[1] = 1.0f;
    v8f c = {};
    v8f d = __builtin_amdgcn_wmma_f32_16x16x4_f32(
        /*neg_a=*/false, a, /*neg_b=*/false, b,
        /*c_mod=*/(short)0, c, /*reuse_a=*/false, /*reuse_b=*/false);
    float s = ((d[0] + d[1]) + (d[2] + d[3])) + ((d[4] + d[5]) + (d[6] + d[7]));
    s += __shfl_xor(s, 16, 32);
    return s;
}

// Kernel 1: per-(channel, slice) partial |a-b| sum + channel validity flag.
// blockIdx.x = ch * SLICES + slice; each block covers SLICE_ELEMS floats.
__global__ __launch_bounds__(NTHREADS)
void genloss_partial_kernel(const float* __restrict__ out_img,
                            const float* __restrict__ tgt_img,
                            float* __restrict__ pSum,
                            unsigned* __restrict__ pValid) {
    const int blk = blockIdx.x;
    const int tid = threadIdx.x;
    const size_t base = (size_t)blk * SLICE_ELEMS + (size_t)tid * 4;

    const float4* o = (const float4*)(out_img + base);
    const float4* t = (const float4*)(tgt_img + base);

    float s    = 0.0f;
    float tmax = 0.0f;   // max |target| seen by this lane; !=0 <=> any nonzero
#pragma unroll
    for (int i = 0; i < VEC_ITERS; ++i) {
        float4 ov = o[i * NTHREADS];   // NTHREADS*16B contiguous per iteration
        float4 tv = t[i * NTHREADS];
        s += fabsf(ov.x - tv.x) + fabsf(ov.y - tv.y)
           + fabsf(ov.z - tv.z) + fabsf(ov.w - tv.w);
        tmax = fmaxf(tmax, fmaxf(fmaxf(fabsf(tv.x), fabsf(tv.y)),
                                 fmaxf(fabsf(tv.z), fabsf(tv.w))));
    }

    // wave32 reductions: sum via WMMA ones-trick, validity via one ballot
    float    ws = wave_sum_wmma(s);
    unsigned wv = (__ballot(tmax != 0.0f) != 0ull) ? 1u : 0u;

    __shared__ float    lsum[NTHREADS / 32];
    __shared__ unsigned lval[NTHREADS / 32];
    const int wave = tid >> 5, lane = tid & 31;
    if (lane == 0) { lsum[wave] = ws; lval[wave] = wv; }
    __syncthreads();

    if (tid == 0) {
        float tot = 0.0f; unsigned vv = 0u;
#pragma unroll
        for (int w = 0; w < NTHREADS / 32; ++w) { tot += lsum[w]; vv |= lval[w]; }
        pSum[blk]   = tot;   // every slot rewritten each call -> no init needed
        pValid[blk] = vv;
    }
}

// Kernel 2: single block. Reduce labels, combine channel partials in a fixed
// order (deterministic), apply valid-channel masking and final loss formula.
__global__ __launch_bounds__(NTHREADS)
void genloss_finalize_kernel(const float* __restrict__ labels, int nLabels,
                             const float* __restrict__ pSum,
                             const unsigned* __restrict__ pValid,
                             const int* __restrict__ epoch_p,
                             float* __restrict__ out) {
    __shared__ float red[NTHREADS];
    __shared__ float mae_s[N_CH];
    __shared__ float val_s[N_CH];
    __shared__ float labelSumSh;

    const int tid = threadIdx.x;

    // ---- adversarial term: sum of labels ----
    float s = 0.0f;
    for (int i = tid; i < nLabels; i += NTHREADS) s += labels[i];
    red[tid] = s;
    __syncthreads();
    for (int step = NTHREADS / 2; step > 0; step >>= 1) {
        if (tid < step) red[tid] += red[tid + step];
        __syncthreads();
    }
    if (tid == 0) labelSumSh = red[0];

    // ---- per-channel MAE + validity (fixed summation order) ----
    if (tid < N_CH) {
        float cs = 0.0f; unsigned vv = 0u;
        for (int sl = 0; sl < SLICES; ++sl) {
            cs += pSum[tid * SLICES + sl];
            vv |= pValid[tid * SLICES + sl];
        }
        mae_s[tid] = cs * (1.0f / (float)CH_ELEMS);
        val_s[tid] = vv ? 1.0f : 0.0f;
    }
    __syncthreads();

    // ---- final combine ----
    if (tid == 0) {
        float image_loss = 0.0f;
        for (int b = 0; b < 16; ++b) {
            float tot = 0.0f, cnt = 0.0f;
            for (int ci = 0; ci < 3; ++ci) {
                tot += mae_s[b * 3 + ci] * val_s[b * 3 + ci];
                cnt += val_s[b * 3 + ci];
            }
            image_loss += (cnt > 0.0f) ? (tot / fmaxf(cnt, 1.0f)) : 0.0f;
        }
        image_loss *= (1.0f / 16.0f);
        const float adv   = -(labelSumSh / (float)nLabels);
        const float epoch = (float)epoch_p[0];
        out[0] = image_loss + 0.01f * adv / (epoch + 1.0f);
    }
}

extern "C" void kernel_launch(void* const* d_in, const int* in_sizes, int n_in,
                              void* d_out, int out_size, void* d_ws, size_t ws_size,
                              hipStream_t stream) {
    const float* labels  = (const float*)d_in[0];  // (16,1,30,30) f32
    const float* out_img = (const float*)d_in[1];  // (16,3,512,512) f32
    const float* tgt_img = (const float*)d_in[2];  // (16,3,512,512) f32
    const int*   epoch   = (const int*)d_in[3];    // scalar int

    float*    pSum   = (float*)d_ws;                                   // 48*32 f32
    unsigned* pValid = (unsigned*)((char*)d_ws + N_CH * SLICES * sizeof(float));

    genloss_partial_kernel<<<N_CH * SLICES, NTHREADS, 0, stream>>>(
        out_img, tgt_img, pSum, pValid);
    genloss_finalize_kernel<<<1, NTHREADS, 0, stream>>>(
        labels, in_sizes[0], pSum, pValid, epoch, (float*)d_out);
}